// PriorMatcher_41618233098698
// MI455X (gfx1250) — compile-verified
//
#include <hip/hip_runtime.h>
#include <stdint.h>

#define T_TGT 32

// ---- CDNA5 async global->LDS copy (inline asm: portable across toolchains) ----
// Each enabled lane copies 16 bytes from the global address in its vaddr to the LDS address in its vdst.
__device__ __forceinline__ void async_copy_b128_to_lds(const void* gptr, void* lds_ptr) {
    uint32_t lds_addr = (uint32_t)(uintptr_t)lds_ptr;   // low 32 bits of flat shared addr == LDS offset
    uint64_t gaddr    = (uint64_t)(uintptr_t)gptr;
    asm volatile("global_load_async_to_lds_b128 %0, %1, off"
                 :: "v"(lds_addr), "v"(gaddr) : "memory");
}
__device__ __forceinline__ void wait_async0() {
    asm volatile("s_wait_asynccnt 0x0" ::: "memory");
}

// Packed argmax key: high 32 = IoU float bits (IoU >= 0 so monotonic as uint),
// low 32 = ~prior_index so u64-max picks highest IoU then SMALLEST prior index
// (matches jnp.argmax first-occurrence tie-breaking).
__global__ void ws_zero(unsigned long long* p, int n) {
    int i = blockIdx.x * blockDim.x + threadIdx.x;
    if (i < n) p[i] = 0ull;
}

__global__ __launch_bounds__(256) void prior_match_phase1(
    const float4* __restrict__ priors, const float4* __restrict__ gt_boxes,
    unsigned long long* __restrict__ ws_target, float* __restrict__ out,
    int P, int B)
{
    const int b   = blockIdx.y;
    const int tid = threadIdx.x;
    const int i   = blockIdx.x * 256 + tid;

    __shared__ float4 s_gt[T_TGT];
    __shared__ unsigned long long s_best[T_TGT];

    if (tid < T_TGT) {                       // wave 0 only: EXEC all-ones within it
        s_best[tid] = 0ull;
        async_copy_b128_to_lds(gt_boxes + (size_t)b * T_TGT + tid, &s_gt[tid]);
        wait_async0();
    }
    __syncthreads();

    const int ii = (i < P) ? i : (P - 1);
    const float4 pr = priors[ii];
    const float hx = pr.z * 0.5f, hy = pr.w * 0.5f;
    const float px1 = pr.x - hx, py1 = pr.y - hy;
    const float px2 = pr.x + hx, py2 = pr.y + hy;
    const float areaP = (px2 - px1) * (py2 - py1);
    const unsigned inv_i = 0xFFFFFFFFu - (unsigned)ii;

    float best_v = -1.0f;
    int   best_t = 0;

    for (int t = 0; t < T_TGT; ++t) {
        const float4 g = s_gt[t];                         // LDS broadcast read
        const float lx = fmaxf(px1, g.x), ly = fmaxf(py1, g.y);
        const float rx = fminf(px2, g.z), ry = fminf(py2, g.w);
        const float w = fmaxf(rx - lx, 0.0f), h = fmaxf(ry - ly, 0.0f);
        const float inter = w * h;
        const float areaG = (g.z - g.x) * (g.w - g.y);
        const float iou = inter / ((areaG + areaP) - inter);
        if (iou > best_v) { best_v = iou; best_t = t; }   // strict > => first-t argmax

        unsigned long long key =
            ((unsigned long long)__float_as_uint(iou) << 32) | (unsigned long long)inv_i;
        if (i >= P) key = 0ull;
        // wave32 max-reduction of packed key
        #pragma unroll
        for (int off = 16; off >= 1; off >>= 1) {
            const unsigned long long o = __shfl_xor(key, off, 32);
            if (o > key) key = o;
        }
        if ((tid & 31) == 0) atomicMax(&s_best[t], key);  // ds_max_u64, 1 per wave
    }
    __syncthreads();
    if (tid < T_TGT)
        atomicMax(&ws_target[(size_t)b * T_TGT + tid], s_best[tid]);  // global_atomic_max_u64

    if (i < P) {
        const size_t pi = (size_t)b * P + i;
        ((int*)out)[pi * 4] = best_t;                 // temp: int bits in loc[...,0] slot
        out[(size_t)B * P * 4 + pi] = best_v;         // temp: float in label slot
    }
}

__global__ __launch_bounds__(256) void prior_match_phase2(
    const float4* __restrict__ priors, const float4* __restrict__ gt_boxes,
    const int* __restrict__ gt_labels, const unsigned long long* __restrict__ ws_target,
    float* __restrict__ out, int P, int B)
{
    const int b   = blockIdx.y;
    const int tid = threadIdx.x;
    const int i   = blockIdx.x * 256 + tid;

    __shared__ float4 s_gt[T_TGT];
    __shared__ int    s_lab[T_TGT];
    __shared__ int    s_bp[T_TGT];

    if (i < P)  // warm L2->L0 path for the temp we read after the barrier
        __builtin_prefetch(&out[(size_t)B * P * 4 + (size_t)b * P + i], 0, 0);

    if (tid < T_TGT) {
        async_copy_b128_to_lds(gt_boxes + (size_t)b * T_TGT + tid, &s_gt[tid]);
        s_lab[tid] = gt_labels[b * T_TGT + tid];
        const unsigned long long k = ws_target[(size_t)b * T_TGT + tid];
        s_bp[tid] = (int)(0xFFFFFFFFu - (unsigned)(k & 0xFFFFFFFFull));
        wait_async0();
    }
    __syncthreads();
    if (i >= P) return;

    const size_t pi = (size_t)b * P + i;
    float* locp = out + pi * 4;
    float* labp = out + (size_t)B * P * 4 + pi;

    float val  = *labp;                    // temp best IoU (read before overwrite)
    int   tidx = ((const int*)locp)[0];    // temp best target idx
    // scatter override: forced matches (last t wins on duplicate best prior)
    #pragma unroll
    for (int t = 0; t < T_TGT; ++t)
        if (s_bp[t] == i) { tidx = t; val = 2.0f; }

    const float4 pr = priors[i];
    const float4 g  = s_gt[tidx];
    const float bcx = (g.x + g.z) * 0.5f;
    const float bcy = (g.y + g.w) * 0.5f;
    const float bw  = g.z - g.x;
    const float bh  = g.w - g.y;

    float4 loc;
    loc.x = ((bcx - pr.x) / pr.z) / 0.1f;
    loc.y = ((bcy - pr.y) / pr.w) / 0.1f;
    loc.z = logf(bw / pr.z) / 0.2f;
    loc.w = logf(bh / pr.w) / 0.2f;
    *(float4*)locp = loc;                  // global_store_b128

    const int lab = (val < 0.5f) ? 0 : s_lab[tidx];
    *labp = (float)lab;
}

extern "C" void kernel_launch(void* const* d_in, const int* in_sizes, int n_in,
                              void* d_out, int out_size, void* d_ws, size_t ws_size,
                              hipStream_t stream) {
    (void)n_in; (void)out_size; (void)ws_size;
    const float4* priors = (const float4*)d_in[0];   // (P,4) f32 xywh
    const float4* gts    = (const float4*)d_in[1];   // (B,T,4) f32 xyxy
    const int*    labels = (const int*)d_in[2];      // (B,T) i32
    const int P  = in_sizes[0] / 4;
    const int BT = in_sizes[2];
    const int B  = BT / T_TGT;

    unsigned long long* ws_target = (unsigned long long*)d_ws;  // B*T packed u64 (4 KB)
    float* out = (float*)d_out;

    ws_zero<<<dim3((BT + 255) / 256), dim3(256), 0, stream>>>(ws_target, BT);
    dim3 grid((P + 255) / 256, B);
    prior_match_phase1<<<grid, dim3(256), 0, stream>>>(priors, gts, ws_target, out, P, B);
    prior_match_phase2<<<grid, dim3(256), 0, stream>>>(priors, gts, labels, ws_target, out, P, B);
}